// HWTV_38706245272444
// MI455X (gfx1250) — compile-verified
//
#include <hip/hip_runtime.h>
#include <math.h>

// ---------------------------------------------------------------------------
// ADMM TV solver on MI455X (gfx1250).
// FFT screened-Poisson solve replaced by the exact real Hartley solve
// (symmetric circulant Laplacian is diagonalized by the cas basis, H*H=N*I):
//   Zhat = (H*R*H) ./ (mu1 + mu2*(w[j]+w[k]));   Z = H*Zhat*H / 65536
// => 4 real 256x256x256 GEMMs / image / iter on V_WMMA_F32_16X16X4_F32.
// GEMM panels are staged with the CDNA5 async global->LDS engine
// (GLOBAL_LOAD_ASYNC_TO_LDS_B128, ASYNCcnt) with LDS double-buffering so DMA
// overlaps WMMA. Elementwise ADMM passes are fused stencils on float4
// (b128 traffic) since the workload floor is the 23.3 TB/s HBM stream.
// ---------------------------------------------------------------------------

#define NIMG   248
#define HW     65536
#define P_ELEMS ((size_t)NIMG * HW)   // 16,252,928
#define ITE    20
#define TWO_PI 6.283185307179586f

typedef __attribute__((ext_vector_type(2))) float v2f;
typedef __attribute__((ext_vector_type(8))) float v8f;

__device__ __forceinline__ float shrinkf(float x, float t) {
    return copysignf(fmaxf(fabsf(x) - t, 0.0f), x);
}

// low 32 bits of a generic pointer to __shared__ = wave-relative LDS offset
__device__ __forceinline__ unsigned lds_off(const void* p) {
    return (unsigned)(unsigned long long)p;
}

// ---------------- init: state = {Z=Y, X=Y, G*=0} ---------------------------
__global__ __launch_bounds__(256)
void k_init_state(const float4* __restrict__ Y4, float4* __restrict__ Z4,
                  float4* __restrict__ X4, float4* __restrict__ G14,
                  float4* __restrict__ G214, float4* __restrict__ G224) {
    const float4 zero = make_float4(0.f, 0.f, 0.f, 0.f);
    for (size_t i = blockIdx.x * blockDim.x + threadIdx.x; i < P_ELEMS / 4;
         i += (size_t)gridDim.x * blockDim.x) {
        float4 y = Y4[i];
        Z4[i] = y; X4[i] = y; G14[i] = zero; G214[i] = zero; G224[i] = zero;
    }
}

// ---------------- init: Hartley matrix, Laplacian eigenvalues, mse=0 -------
__global__ __launch_bounds__(256)
void k_init_const(float* __restrict__ Hm, float* __restrict__ wfreq,
                  float* __restrict__ mse) {
    int tid = blockIdx.x * blockDim.x + threadIdx.x;
    if (tid < HW) {
        int j = tid >> 8, k = tid & 255;
        float ang = (TWO_PI / 256.0f) * (float)((j * k) & 255);
        Hm[tid] = cosf(ang) + sinf(ang);          // cas basis
    }
    if (tid < 256) {
        float ang = (TWO_PI / 256.0f) * (float)tid;
        wfreq[tid] = 2.0f - 2.0f * cosf(ang);     // circular [1,-1] eigenvalues
    }
    if (tid < ITE) mse[tid] = 0.0f;
}

// -------- fused shrink + adjoint stencil -> Uh, Uv, R (float4) -------------
__global__ __launch_bounds__(256)
void k_admm_pre(const float* __restrict__ Z, const float* __restrict__ X,
                const float* __restrict__ G1, const float* __restrict__ G21,
                const float* __restrict__ G22, float* __restrict__ Uh,
                float* __restrict__ Uv, float* __restrict__ R,
                float mu1, float mu2, float lam) {
    const float inv2 = 1.0f / mu2;
    const float thr  = lam * inv2;
    const float4* Z4   = (const float4*)Z;
    const float4* X4   = (const float4*)X;
    const float4* G14  = (const float4*)G1;
    const float4* G214 = (const float4*)G21;
    const float4* G224 = (const float4*)G22;
    for (size_t i4 = blockIdx.x * blockDim.x + threadIdx.x; i4 < P_ELEMS / 4;
         i4 += (size_t)gridDim.x * blockDim.x) {
        size_t i = i4 * 4;
        size_t base = i & ~(size_t)(HW - 1);
        int r = ((int)i >> 8) & 255, c0 = (int)i & 255;
        int rp = (r + 1) & 255, rm = (r + 255) & 255;
        size_t rb = base + ((size_t)r << 8);
        float4 zc  = Z4[i4];
        float4 zn  = Z4[(base + ((size_t)rm << 8) + c0) >> 2];
        float4 zs  = Z4[(base + ((size_t)rp << 8) + c0) >> 2];
        float  zl  = Z[rb + ((c0 + 255) & 255)];      // west halo
        float  zr  = Z[rb + ((c0 + 4) & 255)];        // east halo
        float4 g21 = G214[i4];
        float  g21l = G21[rb + ((c0 + 255) & 255)];
        float4 g22  = G224[i4];
        float4 g22n = G224[(base + ((size_t)rm << 8) + c0) >> 2];
        float4 x  = X4[i4];
        float4 g1 = G14[i4];

        float zca[4]  = {zc.x, zc.y, zc.z, zc.w};
        float zna[4]  = {zn.x, zn.y, zn.z, zn.w};
        float zsa[4]  = {zs.x, zs.y, zs.z, zs.w};
        float g21a[4] = {g21.x, g21.y, g21.z, g21.w};
        float g22a[4] = {g22.x, g22.y, g22.z, g22.w};
        float g22na[4]= {g22n.x, g22n.y, g22n.z, g22n.w};
        float xa[4]   = {x.x, x.y, x.z, x.w};
        float g1a[4]  = {g1.x, g1.y, g1.z, g1.w};
        float uh[4], uv[4], rr[4];
        #pragma unroll
        for (int j = 0; j < 4; ++j) {
            float ze   = (j < 3) ? zca[j + 1] : zr;
            float zw   = (j > 0) ? zca[j - 1] : zl;
            float g21w = (j > 0) ? g21a[j - 1] : g21l;
            float uh_c = shrinkf(ze - zca[j] - g21a[j] * inv2, thr);
            float uh_w = shrinkf(zca[j] - zw - g21w * inv2, thr);
            float uv_c = shrinkf(zsa[j] - zca[j] - g22a[j] * inv2, thr);
            float uv_n = shrinkf(zca[j] - zna[j] - g22na[j] * inv2, thr);
            uh[j] = uh_c;
            uv[j] = uv_c;
            float dhT = (uh_w + g21w * inv2) - (uh_c + g21a[j] * inv2);
            float dvT = (uv_n + g22na[j] * inv2) - (uv_c + g22a[j] * inv2);
            rr[j] = mu1 * xa[j] + g1a[j] + mu2 * (dhT + dvT);
        }
        ((float4*)Uh)[i4] = make_float4(uh[0], uh[1], uh[2], uh[3]);
        ((float4*)Uv)[i4] = make_float4(uv[0], uv[1], uv[2], uv[3]);
        ((float4*)R )[i4] = make_float4(rr[0], rr[1], rr[2], rr[3]);
    }
}

// ---------------- batched WMMA-f32 GEMM with async double-buffering --------
// C = A*B (256x256x256); sA/sB==0 => shared Hartley operand.
// mode 0: plain, mode 1: ./ (mu1+mu2*(w[row]+w[col])), mode 2: * 1/65536
__global__ __launch_bounds__(256)
void k_gemm_h(const float* __restrict__ A, unsigned long long sA,
              const float* __restrict__ B, unsigned long long sB,
              float* __restrict__ C, int mode, float mu1, float mu2,
              const float* __restrict__ wfreq) {
    __shared__ float As[2][64][36];   // padded (rows 144B, 16B-aligned)
    __shared__ float Bs[2][32][68];   // padded (rows 272B, 16B-aligned)
    const int img   = blockIdx.z;
    const float* Ab = A + (size_t)img * sA;
    const float* Bb = B + (size_t)img * sB;
    float*       Cb = C + (size_t)img * HW;
    const int t    = threadIdx.x;
    const int lane = t & 31;                 // wave32
    const int wave = t >> 5;                 // 8 waves / block
    const int hi   = lane >> 4, lo = lane & 15;
    const int mtile = wave >> 1;             // M tile 0..3
    const int nbase = (wave & 1) * 32;       // two 16-wide N tiles per wave
    const int Mblk  = blockIdx.y * 64;
    const int Nblk  = blockIdx.x * 64;

    // async-stage one 64x32 A panel + 32x64 B panel (b128 per lane, ASYNCcnt)
    auto stage = [&](int buf, int k0) {
        const float* baseA = Ab + Mblk * 256 + k0;
        const float* baseB = Bb + k0 * 256 + Nblk;
        #pragma unroll
        for (int q = 0; q < 2; ++q) {
            int f = t + q * 256;                       // 0..511 float4 slots
            int ar = f >> 3, ac = (f & 7) * 4;         // A: 64 x (8 x f4)
            unsigned ldsA = lds_off(&As[buf][ar][ac]);
            unsigned offA = (unsigned)((ar * 256 + ac) * 4);
            asm volatile("global_load_async_to_lds_b128 %0, %1, %2"
                         :: "v"(ldsA), "v"(offA), "s"(baseA) : "memory");
            int br = f >> 4, bc = (f & 15) * 4;        // B: 32 x (16 x f4)
            unsigned ldsB = lds_off(&Bs[buf][br][bc]);
            unsigned offB = (unsigned)((br * 256 + bc) * 4);
            asm volatile("global_load_async_to_lds_b128 %0, %1, %2"
                         :: "v"(ldsB), "v"(offB), "s"(baseB) : "memory");
        }
    };

    v8f acc0 = {}; v8f acc1 = {};
    stage(0, 0);
    for (int chunk = 0; chunk < 8; ++chunk) {
        asm volatile("s_wait_asynccnt 0x0" ::: "memory");
        __syncthreads();                     // current buffer fully in LDS
        int cur = chunk & 1;
        if (chunk < 7) stage(cur ^ 1, (chunk + 1) * 32);  // DMA next chunk
        float (*Asb)[36] = As[cur];
        float (*Bsb)[68] = Bs[cur];
        #pragma unroll
        for (int kk = 0; kk < 32; kk += 4) {
            int ka = kk + 2 * hi;
            v2f a;                 // A 16x4: lanes0-15 K0/K1, lanes16-31 K2/K3
            a.x = Asb[mtile * 16 + lo][ka];
            a.y = Asb[mtile * 16 + lo][ka + 1];
            v2f b0, b1;            // B 4x16 frags for the two N tiles
            b0.x = Bsb[ka][nbase + lo];       b0.y = Bsb[ka + 1][nbase + lo];
            b1.x = Bsb[ka][nbase + 16 + lo];  b1.y = Bsb[ka + 1][nbase + 16 + lo];
            acc0 = __builtin_amdgcn_wmma_f32_16x16x4_f32(
                       false, a, false, b0, (short)0, acc0, false, false);
            acc1 = __builtin_amdgcn_wmma_f32_16x16x4_f32(
                       false, a, false, b1, (short)0, acc1, false, false);
        }
    }
    // C/D layout: vgpr i -> M=i (lanes0-15) / M=i+8 (lanes16-31), N = lo
    const int col0 = Nblk + nbase + lo;
    const float wc0 = (mode == 1) ? wfreq[col0]      : 0.0f;
    const float wc1 = (mode == 1) ? wfreq[col0 + 16] : 0.0f;
    #pragma unroll
    for (int i = 0; i < 8; ++i) {
        int row = Mblk + mtile * 16 + hi * 8 + i;
        float v0 = acc0[i], v1 = acc1[i];
        if (mode == 1) {
            float wr = wfreq[row];
            v0 /= (mu1 + mu2 * (wr + wc0));
            v1 /= (mu1 + mu2 * (wr + wc1));
        } else if (mode == 2) {
            v0 *= (1.0f / 65536.0f);
            v1 *= (1.0f / 65536.0f);
        }
        Cb[row * 256 + col0]      = v0;
        Cb[row * 256 + col0 + 16] = v1;
    }
}

// -------- fused X/G1/G21/G22 update + MSE reduction (float4) ---------------
__global__ __launch_bounds__(256)
void k_admm_post(const float* __restrict__ Z, const float* __restrict__ Y,
                 const float* __restrict__ inW, const float* __restrict__ Uh,
                 const float* __restrict__ Uv, const float* __restrict__ xg,
                 float* __restrict__ X, float* __restrict__ G1,
                 float* __restrict__ G21, float* __restrict__ G22,
                 float* __restrict__ msesum, int it, float mu1, float mu2) {
    const float4* Z4 = (const float4*)Z;
    float acc = 0.0f;
    for (size_t i4 = blockIdx.x * blockDim.x + threadIdx.x; i4 < P_ELEMS / 4;
         i4 += (size_t)gridDim.x * blockDim.x) {
        size_t i = i4 * 4;
        size_t base = i & ~(size_t)(HW - 1);
        int r = ((int)i >> 8) & 255, c0 = (int)i & 255;
        int rp = (r + 1) & 255;
        size_t rb = base + ((size_t)r << 8);
        float4 zc = Z4[i4];
        float4 zs = Z4[(base + ((size_t)rp << 8) + c0) >> 2];
        float  zr = Z[rb + ((c0 + 4) & 255)];         // east halo of new Z
        float4 w  = ((const float4*)inW)[i4];
        float4 y  = ((const float4*)Y)[i4];
        float4 g1 = ((float4*)G1)[i4];
        float4 uh = ((const float4*)Uh)[i4];
        float4 uv = ((const float4*)Uv)[i4];
        float4 g21= ((float4*)G21)[i4];
        float4 g22= ((float4*)G22)[i4];
        float4 xg4= ((const float4*)xg)[i4];

        float zca[4] = {zc.x, zc.y, zc.z, zc.w};
        float zsa[4] = {zs.x, zs.y, zs.z, zs.w};
        float wa[4]  = {w.x, w.y, w.z, w.w};
        float ya[4]  = {y.x, y.y, y.z, y.w};
        float g1a[4] = {g1.x, g1.y, g1.z, g1.w};
        float uha[4] = {uh.x, uh.y, uh.z, uh.w};
        float uva[4] = {uv.x, uv.y, uv.z, uv.w};
        float g21a[4]= {g21.x, g21.y, g21.z, g21.w};
        float g22a[4]= {g22.x, g22.y, g22.z, g22.w};
        float xga[4] = {xg4.x, xg4.y, xg4.z, xg4.w};
        float xo[4], g1o[4], g21o[4], g22o[4];
        #pragma unroll
        for (int j = 0; j < 4; ++j) {
            float ze = (j < 3) ? zca[j + 1] : zr;
            float dh = ze - zca[j];
            float dv = zsa[j] - zca[j];
            float xn = (wa[j] * ya[j] + mu1 * zca[j] - g1a[j]) / (wa[j] + mu1);
            xo[j]  = xn;
            g1o[j] = g1a[j] + mu1 * (xn - zca[j]);
            g21o[j]= g21a[j] + mu2 * (uha[j] - dh);
            g22o[j]= g22a[j] + mu2 * (uva[j] - dv);
            float e = zca[j] - xga[j];
            acc += e * e;
        }
        ((float4*)X  )[i4] = make_float4(xo[0],  xo[1],  xo[2],  xo[3]);
        ((float4*)G1 )[i4] = make_float4(g1o[0], g1o[1], g1o[2], g1o[3]);
        ((float4*)G21)[i4] = make_float4(g21o[0],g21o[1],g21o[2],g21o[3]);
        ((float4*)G22)[i4] = make_float4(g22o[0],g22o[1],g22o[2],g22o[3]);
    }
    __shared__ float red[256];
    red[threadIdx.x] = acc;
    __syncthreads();
    for (int s = 128; s > 0; s >>= 1) {
        if (threadIdx.x < s) red[threadIdx.x] += red[threadIdx.x + s];
        __syncthreads();
    }
    if (threadIdx.x == 0) atomicAdd(msesum + it, red[0]);
}

__global__ __launch_bounds__(256)
void k_final(const float4* __restrict__ Z4, const float* __restrict__ msesum,
             float* __restrict__ out) {
    float4* out4 = (float4*)out;
    for (size_t i = blockIdx.x * blockDim.x + threadIdx.x; i < P_ELEMS / 4;
         i += (size_t)gridDim.x * blockDim.x)
        out4[i] = Z4[i];
    if (blockIdx.x == 0 && threadIdx.x < ITE)
        out[P_ELEMS + threadIdx.x] = msesum[threadIdx.x] * (1.0f / (float)P_ELEMS);
}

extern "C" void kernel_launch(void* const* d_in, const int* in_sizes, int n_in,
                              void* d_out, int out_size, void* d_ws,
                              size_t ws_size, hipStream_t stream) {
    (void)in_sizes; (void)n_in; (void)out_size; (void)ws_size;
    const float* Y   = (const float*)d_in[0];
    const float* inW = (const float*)d_in[1];
    const float* xg  = (const float*)d_in[2];
    // d_in[3] (demo) is reproduced exactly on device as wfreq[j]+wfreq[k].
    float* out = (float*)d_out;
    float* ws  = (float*)d_ws;

    const size_t P = P_ELEMS;
    float *Z   = ws,         *X   = ws + P,     *G1 = ws + 2 * P;
    float *G21 = ws + 3 * P, *G22 = ws + 4 * P;
    float *Uh  = ws + 5 * P, *Uv  = ws + 6 * P;
    float *bA  = ws + 7 * P, *bB  = ws + 8 * P;
    float *Hm  = ws + 9 * P;
    float *wfreq = Hm + HW;
    float *mse   = wfreq + 256;

    dim3 eb(256), eg(4096);
    k_init_state<<<eg, eb, 0, stream>>>((const float4*)Y, (float4*)Z,
                                        (float4*)X, (float4*)G1,
                                        (float4*)G21, (float4*)G22);
    k_init_const<<<(HW + 255) / 256, 256, 0, stream>>>(Hm, wfreq, mse);

    const unsigned long long SB = (unsigned long long)HW;
    double mu1d = 0.1, mu2d = 0.1;          // reference updates mu in float64
    for (int it = 0; it < ITE; ++it) {
        float mu1 = (float)mu1d, mu2 = (float)mu2d;
        k_admm_pre<<<eg, eb, 0, stream>>>(Z, X, G1, G21, G22, Uh, Uv, bA,
                                          mu1, mu2, 0.1f);
        dim3 gg(4, 4, NIMG);
        k_gemm_h<<<gg, eb, 0, stream>>>(Hm, 0ull, bA, SB, bB, 0, mu1, mu2, wfreq);
        k_gemm_h<<<gg, eb, 0, stream>>>(bB, SB, Hm, 0ull, bA, 1, mu1, mu2, wfreq);
        k_gemm_h<<<gg, eb, 0, stream>>>(Hm, 0ull, bA, SB, bB, 0, mu1, mu2, wfreq);
        k_gemm_h<<<gg, eb, 0, stream>>>(bB, SB, Hm, 0ull, Z,  2, mu1, mu2, wfreq);
        k_admm_post<<<eg, eb, 0, stream>>>(Z, Y, inW, Uh, Uv, xg, X, G1, G21,
                                           G22, mse, it, mu1, mu2);
        mu1d *= 1.05; mu2d *= 1.05;
    }
    k_final<<<eg, eb, 0, stream>>>((const float4*)Z, mse, out);
}